// global_average_block_49435073577391
// MI455X (gfx1250) — compile-verified
//
#include <hip/hip_runtime.h>

// Ragged per-segment mean pooling: features [N, D] f32 -> out [B, D] f32.
// Pass 1: HBM-streaming partial sums (b128 nontemporal loads, bandwidth-bound:
//         1 GiB @ 23.3 TB/s ~= 46 us floor; compute is negligible).
// Pass 2: cross-split reduction on the WMMA pipe (ones-A x partials-B trick),
//         full f32 precision via V_WMMA_F32_16X16X4_F32, R specialized at
//         compile time so the loop is 4 branchless back-to-back WMMAs.

typedef __attribute__((ext_vector_type(2))) float v2f;
typedef __attribute__((ext_vector_type(4))) float v4f;
typedef __attribute__((ext_vector_type(8))) float v8f;

// ---------------------------------------------------------------------------
// Pass 1: grid (colTiles, R, B), block 256 (8 waves).
// Each wave owns a 128-float column tile of one row per load (32 lanes x v4f
// = 512 contiguous bytes -> one global_load_b128 per lane, perfectly
// coalesced, non-temporal so the one-shot 1 GiB stream doesn't rinse L2).
// Waves stride rows by 8; 4 independent accumulators keep loads in flight.
// Deterministic LDS tree across the 8 waves at the end.
// ---------------------------------------------------------------------------
__global__ void __launch_bounds__(256)
seg_mean_partial_kernel(const int* __restrict__ lens,
                        const float* __restrict__ feats,
                        float* __restrict__ partials,
                        int Bn, int Dn, int R) {
  const int b     = blockIdx.z;
  const int split = blockIdx.y;

  // Ragged segment start: prefix sum of at most B (=16) ints; uniform/scalar.
  int start = 0;
  for (int i = 0; i < b; ++i) start += lens[i];
  const int len = lens[b];

  // Row range of this split.
  const int chunk = (len + R - 1) / R;
  int r0 = split * chunk;       if (r0 > len) r0 = len;
  int r1 = r0 + chunk;          if (r1 > len) r1 = len;

  const int lane    = threadIdx.x & 31;
  const int rowLane = threadIdx.x >> 5;               // 0..7
  const int colBase = blockIdx.x * 128 + lane * 4;
  const bool colOK  = (colBase + 4 <= Dn);

  v4f a0 = {}, a1 = {}, a2 = {}, a3 = {};

  if (colOK) {
    const float* base   = feats + (size_t)start * (size_t)Dn + colBase;
    const size_t stride = (size_t)Dn;
    int r = r0 + rowLane;
    // Unrolled x4: 4 independent accumulators, rows r, r+8, r+16, r+24.
    for (; r + 24 < r1; r += 32) {
      a0 += __builtin_nontemporal_load((const v4f*)(base + (size_t)(r     ) * stride));
      a1 += __builtin_nontemporal_load((const v4f*)(base + (size_t)(r +  8) * stride));
      a2 += __builtin_nontemporal_load((const v4f*)(base + (size_t)(r + 16) * stride));
      a3 += __builtin_nontemporal_load((const v4f*)(base + (size_t)(r + 24) * stride));
    }
    for (; r < r1; r += 8) {
      a0 += __builtin_nontemporal_load((const v4f*)(base + (size_t)r * stride));
    }
  }
  v4f acc = (a0 + a1) + (a2 + a3);

  // Deterministic cross-wave (rowLane) reduction through LDS.
  __shared__ v4f sm[8][32];
  sm[rowLane][lane] = acc;
  __syncthreads();

  if (rowLane == 0 && colOK) {
    v4f s = sm[0][lane];
    #pragma unroll
    for (int j = 1; j < 8; ++j) s += sm[j][lane];   // fixed order
    *(v4f*)(partials + ((size_t)split * Bn + b) * (size_t)Dn + colBase) = s;
  }
}

// ---------------------------------------------------------------------------
// Pass 2: grid (Dn/16, B), block 32 (one wave). Reduce RR row-split partials
// with the matrix pipe: A = ones(16x4), B = 4x16 tile of partials, so
// D[m,n] = sum_k B[k,n] for every m; accumulate through C over RR/4
// V_WMMA_F32_16X16X4_F32 steps. Ones-A makes the result independent of the
// exact K<->VGPR striping of the B operand. Row M=0 of C (VGPR0, lanes 0-15)
// holds the sums; scale by 1/len.
//
// RR is a compile-time power of two. For RR >= 4 every k-index is statically
// in range -> no per-load predication, fully unrolled branchless loop.
// Column edge handled by clamp + 0/1 multiply (branchless, EXEC stays full).
// ---------------------------------------------------------------------------
template <int RR>
__global__ void __launch_bounds__(32)
seg_mean_final_wmma_kernel(const int* __restrict__ lens,
                           const float* __restrict__ partials,
                           float* __restrict__ out,
                           int Bn, int Dn) {
  const int tile = blockIdx.x;
  const int b    = blockIdx.y;
  const int lane = threadIdx.x;        // 0..31, EXEC all ones (no divergence)
  const int n    = lane & 15;
  const int half = lane >> 4;
  const int col  = tile * 16 + n;
  const bool colOK  = (col < Dn);
  const int  colC   = colOK ? col : (Dn - 1);
  const float cmask = colOK ? 1.0f : 0.0f;

  const size_t plane = (size_t)Bn * (size_t)Dn;       // one split plane
  v8f c = {};
  v2f a = {1.0f, 1.0f};                // all-ones A matrix (both K slots)

  if (RR >= 4) {
    // Fast path: k0 = s + 2*half and k0+1 are provably < RR. Per-lane base
    // pointer hoisted; per-step offset is a uniform scalar multiple of plane.
    const float* lanePtr = partials + (size_t)b * (size_t)Dn + (size_t)colC
                         + (size_t)(2 * half) * plane;
    #pragma unroll
    for (int s = 0; s < RR; s += 4) {
      v2f bv;
      bv.x = lanePtr[(size_t)s * plane]         * cmask;
      bv.y = lanePtr[(size_t)s * plane + plane] * cmask;
      // 8 args: (neg_a, A, neg_b, B, c_mod, C, reuse_a, reuse_b)
      c = __builtin_amdgcn_wmma_f32_16x16x4_f32(
          false, a, false, bv, (short)0, c, false, false);
    }
  } else {
    // RR in {1, 2}: clamp out-of-range k-slots to plane 0 and zero their mask.
    const float* sbase = partials + (size_t)b * (size_t)Dn + (size_t)colC;
    #pragma unroll
    for (int s = 0; s < RR; s += 4) {
      const int k0 = s + 2 * half;
      const int i0 = (k0     < RR) ? k0     : 0;
      const int i1 = (k0 + 1 < RR) ? k0 + 1 : 0;
      const float m0 = (k0     < RR) ? cmask : 0.0f;
      const float m1 = (k0 + 1 < RR) ? cmask : 0.0f;
      v2f bv;
      bv.x = sbase[(size_t)i0 * plane] * m0;
      bv.y = sbase[(size_t)i1 * plane] * m1;
      c = __builtin_amdgcn_wmma_f32_16x16x4_f32(
          false, a, false, bv, (short)0, c, false, false);
    }
  }

  const float inv = 1.0f / (float)lens[b];
  if (colOK && half == 0) {            // C row M=0 lives in c[0], lanes 0..15
    out[(size_t)b * (size_t)Dn + col] = c[0] * inv;
  }
}

// Generic runtime-R fallback (defensive; host only picks powers of two).
__global__ void __launch_bounds__(32)
seg_mean_final_wmma_generic(const int* __restrict__ lens,
                            const float* __restrict__ partials,
                            float* __restrict__ out,
                            int Bn, int Dn, int R) {
  const int tile = blockIdx.x;
  const int b    = blockIdx.y;
  const int lane = threadIdx.x;
  const int n    = lane & 15;
  const int half = lane >> 4;
  const int col  = tile * 16 + n;
  const bool colOK  = (col < Dn);
  const int  colC   = colOK ? col : (Dn - 1);
  const float cmask = colOK ? 1.0f : 0.0f;

  const size_t plane = (size_t)Bn * (size_t)Dn;
  const float* sbase = partials + (size_t)b * (size_t)Dn + (size_t)colC;

  v8f c = {};
  v2f a = {1.0f, 1.0f};
  for (int s = 0; s < R; s += 4) {
    const int k0 = s + 2 * half;
    const int i0 = (k0     < R) ? k0     : 0;
    const int i1 = (k0 + 1 < R) ? k0 + 1 : 0;
    const float m0 = (k0     < R) ? cmask : 0.0f;
    const float m1 = (k0 + 1 < R) ? cmask : 0.0f;
    v2f bv;
    bv.x = sbase[(size_t)i0 * plane] * m0;
    bv.y = sbase[(size_t)i1 * plane] * m1;
    c = __builtin_amdgcn_wmma_f32_16x16x4_f32(
        false, a, false, bv, (short)0, c, false, false);
  }

  const float inv = 1.0f / (float)lens[b];
  if (colOK && half == 0) {
    out[(size_t)b * (size_t)Dn + col] = c[0] * inv;
  }
}

// ---------------------------------------------------------------------------
extern "C" void kernel_launch(void* const* d_in, const int* in_sizes, int n_in,
                              void* d_out, int out_size, void* d_ws, size_t ws_size,
                              hipStream_t stream) {
  (void)n_in;
  const int*   lens  = (const int*)d_in[0];     // stack_lengths [B] int32
  const float* feats = (const float*)d_in[1];   // features [N, D] f32
  float*       outp  = (float*)d_out;           // [B, D] f32

  const int Bn = in_sizes[0];
  const int Dn = out_size / Bn;

  // Row-split factor: 16 gives 1024 pass-1 blocks for full-chip streaming;
  // shrink (deterministically) if the workspace can't hold R*B*D partials.
  int R = 16;
  while (R > 1 && (size_t)R * (size_t)Bn * (size_t)Dn * sizeof(float) > ws_size) R >>= 1;

  float* partials = (float*)d_ws;

  const int colTiles = (Dn + 127) / 128;
  dim3 g1(colTiles, R, Bn), b1(256);
  seg_mean_partial_kernel<<<g1, b1, 0, stream>>>(lens, feats, partials, Bn, Dn, R);

  dim3 g2((Dn + 15) / 16, Bn), b2(32);
  switch (R) {
    case 16: seg_mean_final_wmma_kernel<16><<<g2, b2, 0, stream>>>(lens, partials, outp, Bn, Dn); break;
    case  8: seg_mean_final_wmma_kernel< 8><<<g2, b2, 0, stream>>>(lens, partials, outp, Bn, Dn); break;
    case  4: seg_mean_final_wmma_kernel< 4><<<g2, b2, 0, stream>>>(lens, partials, outp, Bn, Dn); break;
    case  2: seg_mean_final_wmma_kernel< 2><<<g2, b2, 0, stream>>>(lens, partials, outp, Bn, Dn); break;
    case  1: seg_mean_final_wmma_kernel< 1><<<g2, b2, 0, stream>>>(lens, partials, outp, Bn, Dn); break;
    default: seg_mean_final_wmma_generic<<<g2, b2, 0, stream>>>(lens, partials, outp, Bn, Dn, R); break;
  }
}